// Conv_39633958208177
// MI455X (gfx1250) — compile-verified
//
#include <hip/hip_runtime.h>
#include <hip/hip_bf16.h>

// ---------------------------------------------------------------------------
// Problem constants (match reference)
// ---------------------------------------------------------------------------
#define NNODES 20000
#define NEDGES 640000
#define F_IN   64
#define HID    128
#define NCLS   10
#define NGRAPH 128

typedef __attribute__((ext_vector_type(2))) float v2f;
typedef __attribute__((ext_vector_type(8))) float v8f;

// ---------------------------------------------------------------------------
// fp32 atomic add that should lower to global_atomic_add_f32 (L2-resident RMW)
// ---------------------------------------------------------------------------
__device__ __forceinline__ void fatomic_add(float* p, float v) {
#if defined(__HIP_DEVICE_COMPILE__)
    unsafeAtomicAdd(p, v);
#else
    atomicAdd(p, v);
#endif
}

// ---------------------------------------------------------------------------
// Edge scatter-add: agg[dst[e], :] += feat[src[e], :]
// One wave (32 lanes) per edge; lanes stride features -> coalesced loads.
// blockDim = (32, EDGES_PER_BLOCK)
// ---------------------------------------------------------------------------
#define EDGES_PER_BLOCK 8
__global__ void scatter_add_kernel(const float* __restrict__ feat,
                                   const int*   __restrict__ srcIdx,
                                   const int*   __restrict__ dstIdx,
                                   float*       __restrict__ agg,
                                   int E, int F) {
    int e = blockIdx.x * EDGES_PER_BLOCK + threadIdx.y;
    if (e >= E) return;
    int lane = threadIdx.x;                 // 0..31 (wave32)
    int s = srcIdx[e];
    int d = dstIdx[e];
    const float* __restrict__ sp = feat + (long long)s * F;
    float*       __restrict__ dp = agg  + (long long)d * F;
    for (int f = lane; f < F; f += 32) {
        fatomic_add(dp + f, sp[f]);
    }
}

// ---------------------------------------------------------------------------
// Node pooling: pooled[batch[n], :] += h[n, :]   (F == HID)
// One wave per node.
// ---------------------------------------------------------------------------
__global__ void pool_kernel(const float* __restrict__ h,
                            const int*   __restrict__ batch,
                            float*       __restrict__ pooled,
                            int n_nodes) {
    int n = blockIdx.x * EDGES_PER_BLOCK + threadIdx.y;
    if (n >= n_nodes) return;
    int lane = threadIdx.x;
    int g = batch[n];
    const float* __restrict__ sp = h + (long long)n * HID;
    float*       __restrict__ dp = pooled + (long long)g * HID;
    for (int f = lane; f < HID; f += 32) {
        fatomic_add(dp + f, sp[f]);
    }
}

// ---------------------------------------------------------------------------
// Fused dual-GEMM + bias + optional ReLU using V_WMMA_F32_16X16X4_F32:
//   out = act( A1 @ W1 + A2 @ W2 + bias )
// A1,A2: [M,K] row-major; W1,W2: [K,Nout] row-major; out: [M,Nout].
// One wave computes one 16x16 output tile, accumulating both products into
// the same v8f accumulator across K in steps of 4.
//
// ISA fragment layouts (CDNA5 ISA §7.12.2, wave32):
//   A 16x4:  lanes 0-15 -> M=lane,   VGPR0=K+0, VGPR1=K+1
//            lanes16-31 -> M=lane-16,VGPR0=K+2, VGPR1=K+3
//   B 4x16:  lanes 0-15 -> N=lane,   VGPR0=K+0, VGPR1=K+1
//            lanes16-31 -> N=lane-16,VGPR0=K+2, VGPR1=K+3
//   C/D 16x16: VGPR v: lanes 0-15 -> (M=v,   N=lane)
//                      lanes16-31 -> (M=v+8, N=lane-16)
// grid = (M/16, Nout/16), block = 32 (EXEC all-ones, as WMMA requires)
// ---------------------------------------------------------------------------
__global__ void gemm_dual_relu_kernel(const float* __restrict__ A1,
                                      const float* __restrict__ W1,
                                      const float* __restrict__ A2,
                                      const float* __restrict__ W2,
                                      const float* __restrict__ bias,
                                      float*       __restrict__ out,
                                      int M, int K, int Nout, int do_relu) {
    const int lane = threadIdx.x;       // 0..31
    const int half = lane >> 4;         // 0: lanes 0-15, 1: lanes 16-31
    const int l    = lane & 15;
    const int rowA = blockIdx.x * 16 + l;
    const int colB = blockIdx.y * 16 + l;

    v8f acc = {};

    // --- A1 @ W1 ---
    for (int k = 0; k < K; k += 4) {
        const int ka = k + half * 2;
        v2f a, b;
        const float* ap = A1 + rowA * K + ka;
        a.x = ap[0];
        a.y = ap[1];
        b.x = W1[ka * Nout + colB];
        b.y = W1[(ka + 1) * Nout + colB];
        acc = __builtin_amdgcn_wmma_f32_16x16x4_f32(
            /*neg_a=*/false, a, /*neg_b=*/false, b,
            /*c_mod=*/(short)0, acc, /*reuse_a=*/false, /*reuse_b=*/false);
    }
    // --- + A2 @ W2 ---
    for (int k = 0; k < K; k += 4) {
        const int ka = k + half * 2;
        v2f a, b;
        const float* ap = A2 + rowA * K + ka;
        a.x = ap[0];
        a.y = ap[1];
        b.x = W2[ka * Nout + colB];
        b.y = W2[(ka + 1) * Nout + colB];
        acc = __builtin_amdgcn_wmma_f32_16x16x4_f32(
            false, a, false, b, (short)0, acc, false, false);
    }

    const float bv = bias[colB];
    #pragma unroll
    for (int v = 0; v < 8; ++v) {
        const int row = blockIdx.x * 16 + v + half * 8;
        float val = acc[v] + bv;
        if (do_relu) val = fmaxf(val, 0.0f);
        out[row * Nout + colB] = val;
    }
}

// ---------------------------------------------------------------------------
// Head: logits = pooled @ w_out + b_out; out = log_softmax(logits)
// One thread per graph (G = 128 = 1 block).
// ---------------------------------------------------------------------------
__global__ void head_kernel(const float* __restrict__ pooled,
                            const float* __restrict__ w_out,   // [HID, NCLS]
                            const float* __restrict__ b_out,   // [NCLS]
                            float*       __restrict__ out) {   // [NGRAPH, NCLS]
    int g = blockIdx.x * blockDim.x + threadIdx.x;
    if (g >= NGRAPH) return;
    const float* __restrict__ p = pooled + g * HID;
    float logit[NCLS];
    #pragma unroll
    for (int c = 0; c < NCLS; ++c) logit[c] = b_out[c];
    for (int k = 0; k < HID; ++k) {
        const float pv = p[k];
        const float* wr = w_out + k * NCLS;
        #pragma unroll
        for (int c = 0; c < NCLS; ++c) logit[c] = fmaf(pv, wr[c], logit[c]);
    }
    float m = logit[0];
    #pragma unroll
    for (int c = 1; c < NCLS; ++c) m = fmaxf(m, logit[c]);
    float s = 0.0f;
    #pragma unroll
    for (int c = 0; c < NCLS; ++c) s += __expf(logit[c] - m);
    const float lse = m + __logf(s);
    #pragma unroll
    for (int c = 0; c < NCLS; ++c) out[g * NCLS + c] = logit[c] - lse;
}

// ---------------------------------------------------------------------------
// Launch
// ---------------------------------------------------------------------------
extern "C" void kernel_launch(void* const* d_in, const int* in_sizes, int n_in,
                              void* d_out, int out_size, void* d_ws, size_t ws_size,
                              hipStream_t stream) {
    // Inputs in setup_inputs() dict order:
    const float* x        = (const float*)d_in[0];    // [N, F_IN]
    const int*   edge_idx = (const int*)  d_in[1];    // [2, E]
    const int*   batch    = (const int*)  d_in[2];    // [N]
    const float* w1_rel   = (const float*)d_in[3];    // [64,128]
    const float* b1       = (const float*)d_in[4];
    const float* w1_root  = (const float*)d_in[5];
    const float* w2_rel   = (const float*)d_in[6];    // [128,128]
    const float* b2       = (const float*)d_in[7];
    const float* w2_root  = (const float*)d_in[8];
    const float* w3_rel   = (const float*)d_in[9];
    const float* b3       = (const float*)d_in[10];
    const float* w3_root  = (const float*)d_in[11];
    const float* w_out    = (const float*)d_in[12];   // [128,10]
    const float* b_out    = (const float*)d_in[13];

    const int* srcIdx = edge_idx;            // row 0
    const int* dstIdx = edge_idx + NEDGES;   // row 1

    // Workspace carve-up (floats):
    float* agg    = (float*)d_ws;                    // [N, 128] (only N*64 used in layer 1)
    float* h1     = agg    + (size_t)NNODES * HID;   // [N, 128]
    float* h2     = h1     + (size_t)NNODES * HID;   // [N, 128]
    float* h3     = h2     + (size_t)NNODES * HID;   // [N, 128]
    float* pooled = h3     + (size_t)NNODES * HID;   // [G, 128]

    const dim3 scatBlk(32, EDGES_PER_BLOCK);
    const int  scatGrd = (NEDGES + EDGES_PER_BLOCK - 1) / EDGES_PER_BLOCK;
    const dim3 gemmGrd(NNODES / 16, HID / 16);       // 20000/16 = 1250 exact

    // ---- Layer 1 (K = F_IN = 64) ----
    hipMemsetAsync(agg, 0, (size_t)NNODES * F_IN * sizeof(float), stream);
    scatter_add_kernel<<<scatGrd, scatBlk, 0, stream>>>(x, srcIdx, dstIdx, agg, NEDGES, F_IN);
    gemm_dual_relu_kernel<<<gemmGrd, 32, 0, stream>>>(agg, w1_rel, x, w1_root, b1, h1,
                                                      NNODES, F_IN, HID, 1);

    // ---- Layer 2 (K = HID = 128) ----
    hipMemsetAsync(agg, 0, (size_t)NNODES * HID * sizeof(float), stream);
    scatter_add_kernel<<<scatGrd, scatBlk, 0, stream>>>(h1, srcIdx, dstIdx, agg, NEDGES, HID);
    gemm_dual_relu_kernel<<<gemmGrd, 32, 0, stream>>>(agg, w2_rel, h1, w2_root, b2, h2,
                                                      NNODES, HID, HID, 1);

    // ---- Layer 3 ----
    hipMemsetAsync(agg, 0, (size_t)NNODES * HID * sizeof(float), stream);
    scatter_add_kernel<<<scatGrd, scatBlk, 0, stream>>>(h2, srcIdx, dstIdx, agg, NEDGES, HID);
    gemm_dual_relu_kernel<<<gemmGrd, 32, 0, stream>>>(agg, w3_rel, h2, w3_root, b3, h3,
                                                      NNODES, HID, HID, 1);

    // ---- Global add-pool ----
    hipMemsetAsync(pooled, 0, (size_t)NGRAPH * HID * sizeof(float), stream);
    const int poolGrd = (NNODES + EDGES_PER_BLOCK - 1) / EDGES_PER_BLOCK;
    pool_kernel<<<poolGrd, scatBlk, 0, stream>>>(h3, batch, pooled, NNODES);

    // ---- Head: linear + log_softmax ----
    head_kernel<<<1, NGRAPH, 0, stream>>>(pooled, w_out, b_out, (float*)d_out);
}